// PatchManifoldLoss_41472204210221
// MI455X (gfx1250) — compile-verified
//
#include <hip/hip_runtime.h>
#include <hip/hip_bf16.h>
#include <math.h>

typedef __attribute__((ext_vector_type(2))) float v2f;
typedef __attribute__((ext_vector_type(8))) float v8f;

#define B_    8
#define C_    19
#define H_    512
#define W_    512
#define SPLIT_ 4
#define HP    12                   // patch height  (512/4/10)
#define WP    12                   // patch width
#define NPAT  16                   // SPLIT^2
#define NN    (B_*C_*HP*WP)        // 21888 elements per patch row
#define SCALEF (512.0f/48.0f)      // resize scale (out 48 = SPLIT*HP)

// ---------------------------------------------------------------------------
// Kernel 1: fused bilinear resample (gather, antialias=False) + patch-flatten.
// One thread per output element of P[2][16][NN]; reads only the 2x2 input
// neighborhood of its sample point (all samples interior -> no clamping).
// ---------------------------------------------------------------------------
__global__ __launch_bounds__(256) void resample_patch_kernel(
    const float* __restrict__ in0, const float* __restrict__ in1,
    float* __restrict__ P) {
  int idx = blockIdx.x * 256 + threadIdx.x;
  const int total = 2 * NPAT * NN;
  if (idx >= total) return;

  int t   = idx / (NPAT * NN);
  int rem = idx - t * (NPAT * NN);
  int p   = rem / NN;
  int n   = rem - p * NN;

  int x  = n % WP;
  int y  = (n / WP) % HP;
  int ch = (n / (WP * HP)) % C_;
  int b  = n / (WP * HP * C_);

  int i = p / SPLIT_;            // patch grid row
  int j = p % SPLIT_;            // patch grid col
  int R  = i * HP + y;           // row in 48x48 resized image
  int Cc = j * WP + x;           // col in 48x48 resized image

  float ty = (R  + 0.5f) * SCALEF - 0.5f;
  float tx = (Cc + 0.5f) * SCALEF - 0.5f;
  int r0 = (int)floorf(ty);
  int c0 = (int)floorf(tx);
  float fy = ty - (float)r0;
  float fx = tx - (float)c0;

  const float* src  = (t == 0 ? in0 : in1) + (b * C_ + ch) * (H_ * W_);
  const float* row0 = src + r0 * W_ + c0;
  const float* row1 = row0 + W_;
  float v00 = row0[0], v01 = row0[1];
  float v10 = row1[0], v11 = row1[1];
  float top = v00 + fx * (v01 - v00);
  float bot = v10 + fx * (v11 - v10);
  P[idx] = top + fy * (bot - top);
}

// ---------------------------------------------------------------------------
// Kernel 2: map_input via WMMA Gram matrix.
// One workgroup, 8 waves. Each wave accumulates a partial G = P * P^T over a
// K-slice of 21888/8 = 2736 (684 chained v_wmma_f32_16x16x4_f32). Because G
// is symmetric, the 16x4 A fragment equals the 4x16 B fragment bit-for-bit:
//   lane<16 : (P[lane][k+0], P[lane][k+1])
//   lane>=16: (P[lane-16][k+2], P[lane-16][k+3])
// Partials are spilled to LDS using the C/D layout (M = v + 8*(lane>=16),
// N = lane&15) and reduced deterministically; then
//   map_input[r,c] = (G[r,r] + G[c,c] - 2 G[r,c]) / N.
// ---------------------------------------------------------------------------
__global__ __launch_bounds__(256) void gram_wmma_kernel(
    const float* __restrict__ Pi, float* __restrict__ map_in) {
  __shared__ float sG[8 * 256];
  const int tid  = threadIdx.x;
  const int wave = tid >> 5;
  const int lane = tid & 31;
  const int row  = lane & 15;
  const int koff = (lane >> 4) * 2;

  const int kcount = NN / 8;        // 2736, multiple of 4
  const int kbase  = wave * kcount;

  v8f acc = {0.f, 0.f, 0.f, 0.f, 0.f, 0.f, 0.f, 0.f};
  const float* prow = Pi + row * NN;
  for (int k = kbase; k < kbase + kcount; k += 4) {
    v2f frag = *(const v2f*)(prow + k + koff);   // 8B-aligned: NN%4==0, k%4==0
    acc = __builtin_amdgcn_wmma_f32_16x16x4_f32(
        /*neg_a=*/false, frag, /*neg_b=*/false, frag,
        /*c_mod=*/(short)0, acc, /*reuse_a=*/false, /*reuse_b=*/false);
  }

  const int hi   = lane >> 4;
  const int ncol = lane & 15;
#pragma unroll
  for (int v = 0; v < 8; ++v)
    sG[wave * 256 + (v + 8 * hi) * 16 + ncol] = acc[v];
  __syncthreads();

  float g = 0.f;
#pragma unroll
  for (int w = 0; w < 8; ++w) g += sG[w * 256 + tid];
  __syncthreads();
  sG[tid] = g;                       // full 16x16 Gram in sG[0..255]
  __syncthreads();

  int r = tid >> 4, c = tid & 15;
  map_in[tid] = (sG[r * 17] + sG[c * 17] - 2.f * sG[r * 16 + c]) * (1.0f / NN);
}

// ---------------------------------------------------------------------------
// Kernel 3: map_target[r,c] = mean(exp(-(Pt[c]-Pt[r])^2)).
// One block per (r,c) pair, 256-thread strided reduce (deterministic, no
// atomics). Diagonal naturally yields 1.0, matching the reference.
// ---------------------------------------------------------------------------
__global__ __launch_bounds__(256) void target_map_kernel(
    const float* __restrict__ Pt, float* __restrict__ map_tg) {
  __shared__ float red[256];
  const int p = blockIdx.x;
  const int r = p >> 4, c = p & 15;
  const float* pr = Pt + r * NN;
  const float* pc = Pt + c * NN;
  float s = 0.f;
  for (int n = threadIdx.x; n < NN; n += 256) {
    float d = pc[n] - pr[n];
    s += __expf(-d * d);
  }
  red[threadIdx.x] = s;
  __syncthreads();
  for (int st = 128; st > 0; st >>= 1) {
    if (threadIdx.x < st) red[threadIdx.x] += red[threadIdx.x + st];
    __syncthreads();
  }
  if (threadIdx.x == 0) map_tg[p] = red[0] * (1.0f / NN);
}

// ---------------------------------------------------------------------------
// Kernel 4: per-row zero the 2 smallest map_target entries (stable double
// argmin == stable argsort[:2]), dot with map_input, mean over 256, *ALPHA.
// One wave; threads 0..15 each own a row.
// ---------------------------------------------------------------------------
__global__ void finalize_kernel(const float* __restrict__ map_in,
                                const float* __restrict__ map_tg,
                                float* __restrict__ out) {
  __shared__ float part[16];
  int r = threadIdx.x;
  if (r < 16) {
    float mt[16];
#pragma unroll
    for (int j = 0; j < 16; ++j) mt[j] = map_tg[r * 16 + j];
    int i1 = 0; float v1 = mt[0];
#pragma unroll
    for (int j = 1; j < 16; ++j) if (mt[j] < v1) { v1 = mt[j]; i1 = j; }
    int i2 = -1; float v2 = 3.402823466e38f;
#pragma unroll
    for (int j = 0; j < 16; ++j)
      if (j != i1 && mt[j] < v2) { v2 = mt[j]; i2 = j; }
    float s = 0.f;
#pragma unroll
    for (int j = 0; j < 16; ++j)
      if (j != i1 && j != i2) s += map_in[r * 16 + j] * mt[j];
    part[r] = s;
  }
  __syncthreads();
  if (r == 0) {
    float t = 0.f;
#pragma unroll
    for (int j = 0; j < 16; ++j) t += part[j];
    out[0] = t * (1.0f / 256.0f);   // ALPHA == 1.0
  }
}

// ---------------------------------------------------------------------------
extern "C" void kernel_launch(void* const* d_in, const int* in_sizes, int n_in,
                              void* d_out, int out_size, void* d_ws, size_t ws_size,
                              hipStream_t stream) {
  const float* inputs  = (const float*)d_in[0];
  const float* targets = (const float*)d_in[1];
  float* ws  = (float*)d_ws;
  float* P   = ws;                         // [2][16][NN]  (Pi then Pt)
  float* Pi  = P;
  float* Pt  = P + NPAT * NN;
  float* Min = P + 2 * NPAT * NN;          // 256 floats
  float* Mtg = Min + 256;                  // 256 floats
  float* out = (float*)d_out;

  const int total  = 2 * NPAT * NN;        // 700416
  const int blocks = (total + 255) / 256;

  resample_patch_kernel<<<blocks, 256, 0, stream>>>(inputs, targets, P);
  gram_wmma_kernel<<<1, 256, 0, stream>>>(Pi, Min);
  target_map_kernel<<<256, 256, 0, stream>>>(Pt, Mtg);
  finalize_kernel<<<1, 32, 0, stream>>>(Min, Mtg, out);
}